// Net_27865747816552
// MI455X (gfx1250) — compile-verified
//
#include <hip/hip_runtime.h>
#include <hip/hip_bf16.h>
#include <math.h>

typedef float v2f __attribute__((ext_vector_type(2)));
typedef float v8f __attribute__((ext_vector_type(8)));

#define CIN 512
#define H1  16
#define H2  40
#define H2P 48   // 3 WMMA n-tiles of 16

// ---------------------------------------------------------------- gcn_norm
__global__ void k_deg_init(float* __restrict__ deg, int n) {
    int i = blockIdx.x * blockDim.x + threadIdx.x;
    if (i < n) deg[i] = 1.0f;                       // self-loop weight
}

__global__ void k_deg_edges(const int* __restrict__ dst, const float* __restrict__ w,
                            float* __restrict__ deg, int e) {
    int i = blockIdx.x * blockDim.x + threadIdx.x;
    if (i < e) atomicAdd(&deg[dst[i]], w[i]);
}

__global__ void k_dis(const float* __restrict__ deg, float* __restrict__ dis,
                      float* __restrict__ selfn, int n) {
    int i = blockIdx.x * blockDim.x + threadIdx.x;
    if (i < n) {
        float d = deg[i];
        float r = (d > 0.0f) ? rsqrtf(d) : 0.0f;
        dis[i]   = r;
        selfn[i] = r * r;                           // norm of the self-loop edge
    }
}

__global__ void k_norm(const int* __restrict__ src, const int* __restrict__ dst,
                       const float* __restrict__ w, const float* __restrict__ dis,
                       float* __restrict__ norm, int e) {
    int i = blockIdx.x * blockDim.x + threadIdx.x;
    if (i < e) norm[i] = dis[src[i]] * w[i] * dis[dst[i]];
}

// ---------------------------------------------------------------- GEMM1: xw1 = x @ W1  (K=512, N=16)
__global__ __launch_bounds__(256) void k_gemm1(const float* __restrict__ x,
                                               const float* __restrict__ W1,
                                               float* __restrict__ xw1,
                                               int nrows, int ntiles) {
    __shared__ float sW[CIN * H1];                  // 32 KB of 320 KB LDS
    for (int i = threadIdx.x; i < CIN * H1; i += 256) sW[i] = W1[i];
    __syncthreads();

    const int lane = threadIdx.x & 31;
    const int wave = threadIdx.x >> 5;
    const int tile = blockIdx.x * 8 + wave;
    if (tile >= ntiles) return;

    const int m  = lane & 15;                       // A row / B,C col within tile
    const int kh = (lane >> 4) << 1;                // 0 or 2 (A/B K sub-phase)
    int arowi = tile * 16 + m; if (arowi >= nrows) arowi = nrows - 1;
    const float* __restrict__ arow = x + (size_t)arowi * CIN + kh;

    v8f acc = {};
    #pragma unroll 8
    for (int k = 0; k < CIN; k += 4) {
        v2f a; a.x = arow[k];                       // K = kh+0   (b64 load)
               a.y = arow[k + 1];                   // K = kh+1
        v2f b; b.x = sW[(k + kh) * H1 + m];         // B[K][n]
               b.y = sW[(k + kh + 1) * H1 + m];
        acc = __builtin_amdgcn_wmma_f32_16x16x4_f32(
                  false, a, false, b, (short)0, acc, false, false);
    }

    const int mb = tile * 16 + ((lane >> 4) << 3);  // C/D: VGPR v -> row v (+8 for hi lanes)
    if (tile * 16 + 16 <= nrows) {                  // wave-uniform fast path: no exec juggling
        #pragma unroll
        for (int v = 0; v < 8; ++v)
            xw1[(size_t)(mb + v) * H1 + m] = acc[v];
    } else {
        #pragma unroll
        for (int v = 0; v < 8; ++v) {
            int r = mb + v;
            if (r < nrows) xw1[(size_t)r * H1 + m] = acc[v];
        }
    }
}

// ---------------------------------------------------------------- layer-1 scatter
__global__ void k_sloop16(const float* __restrict__ xw, const float* __restrict__ selfn,
                          float* __restrict__ out, int tot) {   // tot = N*16
    int t = blockIdx.x * blockDim.x + threadIdx.x;
    if (t < tot) out[t] = xw[t] * selfn[t >> 4];
}

__global__ void k_edges16(const int* __restrict__ src, const int* __restrict__ dst,
                          const float* __restrict__ norm, const float* __restrict__ xw,
                          float* __restrict__ out, int tot) {   // tot = E*16
    int t = blockIdx.x * blockDim.x + threadIdx.x;
    if (t < tot) {
        int e = t >> 4, j = t & 15;
        atomicAdd(&out[dst[e] * H1 + j], xw[src[e] * H1 + j] * norm[e]);
    }
}

__global__ void k_relu_bias(float* __restrict__ h, const float* __restrict__ b, int tot) {
    int t = blockIdx.x * blockDim.x + threadIdx.x;
    if (t < tot) {
        float v = h[t] + b[t & 15];
        h[t] = v > 0.0f ? v : 0.0f;
    }
}

// ---------------------------------------------------------------- GEMM2: xw2 = h @ W2  (K=16, N=40->48)
__global__ __launch_bounds__(256) void k_gemm2(const float* __restrict__ h,
                                               const float* __restrict__ W2,
                                               float* __restrict__ xw2,
                                               int nrows, int ntiles) {
    __shared__ float sW[H1 * H2P];                  // zero-padded to 48 cols
    for (int i = threadIdx.x; i < H1 * H2P; i += 256) {
        int r = i / H2P, c = i % H2P;
        sW[i] = (c < H2) ? W2[r * H2 + c] : 0.0f;
    }
    __syncthreads();

    const int lane = threadIdx.x & 31;
    const int wave = threadIdx.x >> 5;
    const int tile = blockIdx.x * 8 + wave;
    if (tile >= ntiles) return;

    const int m  = lane & 15;
    const int kh = (lane >> 4) << 1;
    int arowi = tile * 16 + m; if (arowi >= nrows) arowi = nrows - 1;
    const float* __restrict__ arow = h + (size_t)arowi * H1 + kh;
    const int mb = tile * 16 + ((lane >> 4) << 3);
    const bool fulltile = (tile * 16 + 16 <= nrows);  // wave-uniform

    v2f a[4];
    #pragma unroll
    for (int kk = 0; kk < 4; ++kk) {                // preload A (reused for all 3 n-tiles)
        a[kk].x = arow[kk * 4];
        a[kk].y = arow[kk * 4 + 1];
    }

    for (int nt = 0; nt < 3; ++nt) {
        v8f acc = {};
        #pragma unroll
        for (int kk = 0; kk < 4; ++kk) {
            int k = kk * 4;
            v2f b; b.x = sW[(k + kh) * H2P + nt * 16 + m];
                   b.y = sW[(k + kh + 1) * H2P + nt * 16 + m];
            acc = __builtin_amdgcn_wmma_f32_16x16x4_f32(
                      false, a[kk], false, b, (short)0, acc, false, false);
        }
        int col = nt * 16 + m;
        if (col < H2) {
            if (fulltile) {
                #pragma unroll
                for (int v = 0; v < 8; ++v)
                    xw2[(size_t)(mb + v) * H2 + col] = acc[v];
            } else {
                #pragma unroll
                for (int v = 0; v < 8; ++v) {
                    int r = mb + v;
                    if (r < nrows) xw2[(size_t)r * H2 + col] = acc[v];
                }
            }
        }
    }
}

// ---------------------------------------------------------------- layer-2 scatter
__global__ void k_sloop40(const float* __restrict__ xw, const float* __restrict__ selfn,
                          float* __restrict__ out, int tot) {   // tot = N*40
    int t = blockIdx.x * blockDim.x + threadIdx.x;
    if (t < tot) out[t] = xw[t] * selfn[(unsigned)t / 40u];
}

__global__ void k_edges40(const int* __restrict__ src, const int* __restrict__ dst,
                          const float* __restrict__ norm, const float* __restrict__ xw,
                          float* __restrict__ out, int tot) {   // tot = E*40
    int t = blockIdx.x * blockDim.x + threadIdx.x;
    if (t < tot) {
        unsigned e = (unsigned)t / 40u, j = (unsigned)t % 40u;
        atomicAdd(&out[dst[e] * H2 + j], xw[src[e] * H2 + j] * norm[e]);
    }
}

// ---------------------------------------------------------------- bias + log_softmax (in place, one wave per node)
__global__ __launch_bounds__(256) void k_logsoftmax(float* __restrict__ out,
                                                    const float* __restrict__ b2, int n) {
    const int lane = threadIdx.x & 31;
    const int wave = threadIdx.x >> 5;
    const int i = blockIdx.x * 8 + wave;
    if (i >= n) return;
    float* row = out + (size_t)i * H2;
    const int j1 = lane + 32;
    float z0 = (lane < H2) ? row[lane] + b2[lane] : -INFINITY;
    float z1 = (j1   < H2) ? row[j1]   + b2[j1]   : -INFINITY;

    float mx = fmaxf(z0, z1);
    #pragma unroll
    for (int s = 16; s; s >>= 1) mx = fmaxf(mx, __shfl_xor(mx, s, 32));

    float se = ((lane < H2) ? expf(z0 - mx) : 0.0f) + ((j1 < H2) ? expf(z1 - mx) : 0.0f);
    #pragma unroll
    for (int s = 16; s; s >>= 1) se += __shfl_xor(se, s, 32);

    float lse = mx + logf(se);
    if (lane < H2) row[lane] = z0 - lse;
    if (j1   < H2) row[j1]   = z1 - lse;
}

// ---------------------------------------------------------------- driver
extern "C" void kernel_launch(void* const* d_in, const int* in_sizes, int n_in,
                              void* d_out, int out_size, void* d_ws, size_t ws_size,
                              hipStream_t stream) {
    const float* x  = (const float*)d_in[0];
    const int*   ei = (const int*)  d_in[1];   // [2, E] (JAX x64 disabled -> int32)
    const float* ew = (const float*)d_in[2];
    const float* W1 = (const float*)d_in[3];
    const float* b1 = (const float*)d_in[4];
    const float* W2 = (const float*)d_in[5];
    const float* b2 = (const float*)d_in[6];

    const int N = in_sizes[0] / CIN;
    const int E = in_sizes[2];
    const int* src = ei;
    const int* dst = ei + E;

    // workspace carve-out (floats, 256 B aligned slices)
    float* ws = (float*)d_ws;
    size_t off = 0;
    auto carve = [&](size_t count) {
        float* p = ws + off;
        off += (count + 63) & ~(size_t)63;
        return p;
    };
    float* deg   = carve((size_t)N);
    float* dis   = carve((size_t)N);
    float* selfn = carve((size_t)N);
    float* norm  = carve((size_t)E);
    float* xw1   = carve((size_t)N * H1);
    float* h1    = carve((size_t)N * H1);   // accumulator, then relu'd hidden
    float* xw2   = carve((size_t)N * H2);
    float* out2  = (float*)d_out;           // layer-2 accumulator, softmaxed in place

    const int T = 256;
    auto cdiv = [](long long a, long long b) { return (int)((a + b - 1) / b); };

    // ---- gcn_norm (computed once, reused by both layers)
    k_deg_init <<<cdiv(N, T), T, 0, stream>>>(deg, N);
    k_deg_edges<<<cdiv(E, T), T, 0, stream>>>(dst, ew, deg, E);
    k_dis      <<<cdiv(N, T), T, 0, stream>>>(deg, dis, selfn, N);
    k_norm     <<<cdiv(E, T), T, 0, stream>>>(src, dst, ew, dis, norm, E);

    // ---- layer 1
    const int ntiles = (N + 15) / 16;
    k_gemm1    <<<cdiv(ntiles, 8), T, 0, stream>>>(x, W1, xw1, N, ntiles);
    k_sloop16  <<<cdiv((long long)N * H1, T), T, 0, stream>>>(xw1, selfn, h1, N * H1);
    k_edges16  <<<cdiv((long long)E * H1, T), T, 0, stream>>>(src, dst, norm, xw1, h1, E * H1);
    k_relu_bias<<<cdiv((long long)N * H1, T), T, 0, stream>>>(h1, b1, N * H1);

    // ---- layer 2
    k_gemm2    <<<cdiv(ntiles, 8), T, 0, stream>>>(h1, W2, xw2, N, ntiles);
    k_sloop40  <<<cdiv((long long)N * H2, T), T, 0, stream>>>(xw2, selfn, out2, N * H2);
    k_edges40  <<<cdiv((long long)E * H2, T), T, 0, stream>>>(src, dst, norm, xw2, out2, E * H2);

    // ---- bias + log_softmax
    k_logsoftmax<<<cdiv(N, 8), T, 0, stream>>>(out2, b2, N);
}